// GCN_4432406250067
// MI455X (gfx1250) — compile-verified
//
#include <hip/hip_runtime.h>
#include <cstdint>

#define N_NODES 100000
#define N_EDGES 1600000
#define D_IN    128
#define D_HID   64
#define D_OUT   32

typedef __attribute__((ext_vector_type(2))) float v2f;
typedef __attribute__((ext_vector_type(8))) float v8f;

// ---------------------------------------------------------------------------
// Init: deg = 1 (self-loop contribution), zero atomic-accumulation targets.
// ---------------------------------------------------------------------------
__global__ void gcn_init_kernel(float* __restrict__ deg,
                                float* __restrict__ agg1,
                                float* __restrict__ out) {
    int idx = blockIdx.x * blockDim.x + threadIdx.x;
    if (idx < N_NODES * D_HID) agg1[idx] = 0.0f;
    if (idx < N_NODES * D_OUT) out[idx]  = 0.0f;
    if (idx < N_NODES)         deg[idx]  = 1.0f;
}

// ---------------------------------------------------------------------------
// Degree: deg[dst] += 1 per edge (native fp32 global atomic).
// ---------------------------------------------------------------------------
__global__ void gcn_degree_kernel(const int* __restrict__ dst,
                                  float* __restrict__ deg) {
    int e = blockIdx.x * blockDim.x + threadIdx.x;
    if (e < N_EDGES) {
        unsafeAtomicAdd(&deg[dst[e]], 1.0f);
    }
}

// deg -> dinv = rsqrt(deg), in place (deg >= 1 always, self-loop).
__global__ void gcn_rsqrt_kernel(float* __restrict__ deg) {
    int i = blockIdx.x * blockDim.x + threadIdx.x;
    if (i < N_NODES) deg[i] = rsqrtf(deg[i]);
}

// ---------------------------------------------------------------------------
// Dense transform via V_WMMA_F32_16X16X4_F32 (exact fp32 math).
// One wave computes a 16x16 tile of OUT = A[nrows x K] @ W[K x NOUT].
//
// fp32 A-fragment layout (ISA 7.12.2, 32-bit A 16x4):
//   lane l (0..31): row M = l & 15
//   VGPR0 = A[M][k + 2*(l>>4)], VGPR1 = A[M][k + 2*(l>>4) + 1]
// B-fragment (4x16, K x N) mirrored: col N = l & 15,
//   VGPR0 = W[k + 2*(l>>4)][N], VGPR1 = W[k + 2*(l>>4) + 1][N]
// C/D: VGPR r, lane l -> row tile_m*16 + 8*(l>>4) + r, col tile_n*16 + (l&15)
// ---------------------------------------------------------------------------
template <int K, int NOUT>
__global__ void gcn_gemm_wmma_kernel(const float* __restrict__ A,
                                     const float* __restrict__ W,
                                     float* __restrict__ OUT,
                                     int nrows) {
    constexpr int NT = NOUT / 16;                 // column tiles
    const int gw = (blockIdx.x * blockDim.x + threadIdx.x) >> 5;  // global wave
    const int ntiles = (nrows / 16) * NT;
    if (gw >= ntiles) return;                     // wave-uniform: EXEC all-1 inside

    const int tile_m = gw / NT;
    const int tile_n = gw % NT;
    const int lane   = threadIdx.x & 31;
    const int lan16  = lane & 15;
    const int khalf  = lane >> 4;                 // 0 or 1

    const int mrow = tile_m * 16 + lan16;         // A row for this lane
    const int ncol = tile_n * 16 + lan16;         // B/D column for this lane

    const float* arow = A + (size_t)mrow * K;

    v8f acc = {};
#pragma unroll
    for (int k = 0; k < K; k += 4) {
        const int ka = k + 2 * khalf;
        v2f av, bv;
        av.x = arow[ka];
        av.y = arow[ka + 1];
        bv.x = W[(size_t)ka * NOUT + ncol];
        bv.y = W[(size_t)(ka + 1) * NOUT + ncol];
        acc = __builtin_amdgcn_wmma_f32_16x16x4_f32(
            /*neg_a=*/false, av, /*neg_b=*/false, bv,
            /*c_mod=*/(short)0, acc, /*reuse_a=*/false, /*reuse_b=*/false);
    }

    const int mbase = tile_m * 16 + 8 * khalf;
#pragma unroll
    for (int r = 0; r < 8; ++r) {
        OUT[(size_t)(mbase + r) * NOUT + ncol] = acc[r];
    }
}

// ---------------------------------------------------------------------------
// Edge scatter: agg[dst] += h[src] * (dinv[src] * dinv[dst])
// D floats per edge; each thread handles one float4 chunk -> 4 fp32 atomics.
// ---------------------------------------------------------------------------
template <int D>
__global__ void gcn_scatter_kernel(const int* __restrict__ src,
                                   const int* __restrict__ dst,
                                   const float* __restrict__ dinv,
                                   const float* __restrict__ h,
                                   float* __restrict__ agg) {
    constexpr int CH = D / 4;                     // float4 chunks per edge
    const int gid = blockIdx.x * blockDim.x + threadIdx.x;
    if (gid >= N_EDGES * CH) return;
    const int e  = gid / CH;
    const int c4 = (gid % CH) * 4;
    const int s  = src[e];
    const int d  = dst[e];
    const float nrm = dinv[s] * dinv[d];
    const float4 hv = *(const float4*)(h + (size_t)s * D + c4);
    float* p = agg + (size_t)d * D + c4;
    unsafeAtomicAdd(p + 0, hv.x * nrm);
    unsafeAtomicAdd(p + 1, hv.y * nrm);
    unsafeAtomicAdd(p + 2, hv.z * nrm);
    unsafeAtomicAdd(p + 3, hv.w * nrm);
}

// agg = relu(agg + dinv^2 * h + b)   (self-loop folded in; in place)
__global__ void gcn_bias_relu_kernel(const float* __restrict__ h,
                                     const float* __restrict__ dinv,
                                     const float* __restrict__ b,
                                     float* __restrict__ agg) {
    int idx = blockIdx.x * blockDim.x + threadIdx.x;
    if (idx >= N_NODES * D_HID) return;
    const int i = idx >> 6;       // / 64
    const int c = idx & 63;
    const float di = dinv[i];
    float v = agg[idx] + di * di * h[idx] + b[c];
    agg[idx] = v > 0.0f ? v : 0.0f;
}

// out += dinv^2 * h2 + b2   (out already holds edge-scattered sum)
__global__ void gcn_finalize_kernel(const float* __restrict__ h2,
                                    const float* __restrict__ dinv,
                                    const float* __restrict__ b2,
                                    float* __restrict__ out) {
    int idx = blockIdx.x * blockDim.x + threadIdx.x;
    if (idx >= N_NODES * D_OUT) return;
    const int i = idx >> 5;       // / 32
    const int c = idx & 31;
    const float di = dinv[i];
    out[idx] += di * di * h2[idx] + b2[c];
}

// ---------------------------------------------------------------------------
extern "C" void kernel_launch(void* const* d_in, const int* in_sizes, int n_in,
                              void* d_out, int out_size, void* d_ws, size_t ws_size,
                              hipStream_t stream) {
    const float* x    = (const float*)d_in[0];           // [N, 128]
    const int*   ei   = (const int*)d_in[1];             // [2, E]
    const float* W1   = (const float*)d_in[2];           // [128, 64]
    const float* b1   = (const float*)d_in[3];           // [64]
    const float* W2   = (const float*)d_in[4];           // [64, 32]
    const float* b2   = (const float*)d_in[5];           // [32]
    float*       out  = (float*)d_out;                   // [N, 32]

    const int* src = ei;                 // edge_index[0]
    const int* dst = ei + N_EDGES;       // edge_index[1]

    // Workspace carve-up (floats): deg/dinv | H1 | AGG1 | H2  (~64.4 MB total)
    float* deg  = (float*)d_ws;
    float* H1   = deg  + N_NODES;
    float* AGG1 = H1   + (size_t)N_NODES * D_HID;
    float* H2   = AGG1 + (size_t)N_NODES * D_HID;

    const int TB = 256;

    // 1) init deg=1, zero AGG1 and out
    {
        int n = N_NODES * D_HID;
        gcn_init_kernel<<<(n + TB - 1) / TB, TB, 0, stream>>>(deg, AGG1, out);
    }
    // 2) degree accumulation
    gcn_degree_kernel<<<(N_EDGES + TB - 1) / TB, TB, 0, stream>>>(dst, deg);
    // 3) dinv = rsqrt(deg)
    gcn_rsqrt_kernel<<<(N_NODES + TB - 1) / TB, TB, 0, stream>>>(deg);

    // 4) H1 = X @ W1   (wave per 16x16 tile)
    {
        int tiles = (N_NODES / 16) * (D_HID / 16);
        int threads = tiles * 32;
        gcn_gemm_wmma_kernel<D_IN, D_HID>
            <<<(threads + TB - 1) / TB, TB, 0, stream>>>(x, W1, H1, N_NODES);
    }
    // 5) AGG1 += scatter(h1[src] * norm)
    {
        int n = N_EDGES * (D_HID / 4);
        gcn_scatter_kernel<D_HID>
            <<<(n + TB - 1) / TB, TB, 0, stream>>>(src, dst, deg, H1, AGG1);
    }
    // 6) AGG1 = relu(AGG1 + dinv^2*H1 + b1)
    {
        int n = N_NODES * D_HID;
        gcn_bias_relu_kernel<<<(n + TB - 1) / TB, TB, 0, stream>>>(H1, deg, b1, AGG1);
    }
    // 7) H2 = AGG1 @ W2
    {
        int tiles = (N_NODES / 16) * (D_OUT / 16);
        int threads = tiles * 32;
        gcn_gemm_wmma_kernel<D_HID, D_OUT>
            <<<(threads + TB - 1) / TB, TB, 0, stream>>>(AGG1, W2, H2, N_NODES);
    }
    // 8) out += scatter(h2[src] * norm)
    {
        int n = N_EDGES * (D_OUT / 4);
        gcn_scatter_kernel<D_OUT>
            <<<(n + TB - 1) / TB, TB, 0, stream>>>(src, dst, deg, H2, out);
    }
    // 9) out += dinv^2*H2 + b2
    {
        int n = N_NODES * D_OUT;
        gcn_finalize_kernel<<<(n + TB - 1) / TB, TB, 0, stream>>>(H2, deg, b2, out);
    }
}